// S2SDecoder_84799834292451
// MI455X (gfx1250) — compile-verified
//
#include <hip/hip_runtime.h>
#include <math.h>

// Problem constants (from reference setup_inputs)
#define TT 32      // time steps
#define BBATCH 32  // batch
#define SSRC 2048  // encoder length
#define HID 512    // hidden
#define IIN 512    // input feature

typedef __attribute__((ext_vector_type(2))) float v2f;
typedef __attribute__((ext_vector_type(8))) float v8f;

// ---------------------------------------------------------------------------
// Generic batched fp32 WMMA GEMM:
//   C[bat, m, n] = sum_k A[bat, m, k] * B(bat, k, n)  (+ bias[n])
// TRANS_B = true : B supplied as weight W[n, k], row stride ldb (PyTorch W.T)
// TRANS_B = false: B supplied as [k, n], row stride ldb
// Wave tile: 16(M) x 64(N) -> 4 accumulators share each A fragment
// Block: 4 waves -> 16 x 256. K stepped by 4 (v_wmma_f32_16x16x4_f32).
// Requires M%16==0, N%256==0, K%4==0 (all shapes here satisfy this).
// ---------------------------------------------------------------------------
template<bool TRANS_B>
__global__ __launch_bounds__(128)
void gemm_wmma_f32(const float* __restrict__ A, long lda, long sA,
                   const float* __restrict__ Bm, long ldb, long sB,
                   const float* __restrict__ bias,
                   float* __restrict__ C, long ldc, long sC,
                   int M, int N, int K)
{
    const int lane  = threadIdx.x & 31;
    const int wave  = threadIdx.x >> 5;
    const int n0    = (blockIdx.x * (blockDim.x >> 5) + wave) << 6;  // 64-wide per wave
    const int m0    = blockIdx.y << 4;
    const int bat   = blockIdx.z;
    if (n0 >= N || m0 >= M) return;            // wave-uniform: EXEC stays all-ones

    const int idx16 = lane & 15;
    const int khalf = (lane >> 4) << 1;        // 0 or 2: K sub-pair per lane half

    // A fragment base: A[m0+idx16, khalf] (float2 covers K=khalf, khalf+1)
    const float* Ap = A + (long)bat * sA + (long)(m0 + idx16) * lda + khalf;

    const float* Bp[4];
    #pragma unroll
    for (int nn = 0; nn < 4; ++nn) {
        const int nc = n0 + (nn << 4) + idx16;
        if (TRANS_B) Bp[nn] = Bm + (long)bat * sB + (long)nc * ldb + khalf;
        else         Bp[nn] = Bm + (long)bat * sB + (long)khalf * ldb + nc;
    }

    v8f acc[4];
    #pragma unroll
    for (int nn = 0; nn < 4; ++nn) {
        const float bv = bias ? bias[n0 + (nn << 4) + idx16] : 0.0f;
        #pragma unroll
        for (int i = 0; i < 8; ++i) acc[nn][i] = bv;
    }

    #pragma unroll 2
    for (int k = 0; k < K; k += 4) {
        const v2f a = *(const v2f*)(Ap + k);
        #pragma unroll
        for (int nn = 0; nn < 4; ++nn) {
            v2f b;
            if (TRANS_B) {
                b = *(const v2f*)(Bp[nn] + k);
            } else {
                const float* bq = Bp[nn] + (long)k * ldb;
                b[0] = bq[0];
                b[1] = bq[ldb];
            }
            acc[nn] = __builtin_amdgcn_wmma_f32_16x16x4_f32(
                          false, a, false, b, (short)0, acc[nn], false, false);
        }
    }

    // C tile: VGPR j -> row m0 + j + 8*(lane>=16), col n0 + 16*nn + idx16
    float* Cp = C + (long)bat * sC + (long)(m0 + ((lane >> 4) << 3)) * ldc + n0 + idx16;
    #pragma unroll
    for (int nn = 0; nn < 4; ++nn) {
        #pragma unroll
        for (int j = 0; j < 8; ++j) Cp[(long)j * ldc + (nn << 4)] = acc[nn][j];
    }
}

// ---------------------------------------------------------------------------
// One GRU time step, fused: gh = h @ w_hh.T + b_hh for all three gates via
// three interleaved WMMA K-loops sharing the A fragment, then the gate
// nonlinearities applied directly on the accumulator tile.
// grid = (HID/16, BBATCH/16), 1 wave per block.
// ---------------------------------------------------------------------------
__device__ __forceinline__ float sigmoidf_(float x) { return 1.0f / (1.0f + expf(-x)); }

__global__ __launch_bounds__(32)
void gru_step_wmma(const float* __restrict__ gi,   // [B, 3H] rows for this t
                   const float* __restrict__ hin,  // [B, H]
                   const float* __restrict__ whh,  // [3H, H]
                   const float* __restrict__ bhh,  // [3H]
                   float* __restrict__ hout,       // [B, H]
                   float* __restrict__ outt,       // out + t*B*H
                   float* __restrict__ hlast)      // d_out hidden slot or nullptr
{
    const int lane  = threadIdx.x & 31;
    const int idx16 = lane & 15;
    const int khalf = (lane >> 4) << 1;
    const int n0    = blockIdx.x << 4;   // column tile within H
    const int m0    = blockIdx.y << 4;   // batch-row tile

    const float* Ap = hin + (long)(m0 + idx16) * HID + khalf;
    const float* Br = whh + (long)(0 * HID + n0 + idx16) * HID + khalf;  // r gate rows
    const float* Bz = whh + (long)(1 * HID + n0 + idx16) * HID + khalf;  // z gate rows
    const float* Bn = whh + (long)(2 * HID + n0 + idx16) * HID + khalf;  // n gate rows

    v8f ar, az, an;
    {
        const float br = bhh[0 * HID + n0 + idx16];
        const float bz = bhh[1 * HID + n0 + idx16];
        const float bn = bhh[2 * HID + n0 + idx16];
        #pragma unroll
        for (int i = 0; i < 8; ++i) { ar[i] = br; az[i] = bz; an[i] = bn; }
    }

    #pragma unroll 4
    for (int k = 0; k < HID; k += 4) {
        v2f a  = *(const v2f*)(Ap + k);
        v2f br = *(const v2f*)(Br + k);
        v2f bz = *(const v2f*)(Bz + k);
        v2f bn = *(const v2f*)(Bn + k);
        ar = __builtin_amdgcn_wmma_f32_16x16x4_f32(false, a, false, br, (short)0, ar, false, false);
        az = __builtin_amdgcn_wmma_f32_16x16x4_f32(false, a, false, bz, (short)0, az, false, false);
        an = __builtin_amdgcn_wmma_f32_16x16x4_f32(false, a, false, bn, (short)0, an, false, false);
    }

    const int mbase = m0 + ((lane >> 4) << 3);
    const int col   = n0 + idx16;
    #pragma unroll
    for (int j = 0; j < 8; ++j) {
        const int   m   = mbase + j;
        const float ir  = gi[(long)m * (3 * HID) + 0 * HID + col];
        const float iz  = gi[(long)m * (3 * HID) + 1 * HID + col];
        const float inn = gi[(long)m * (3 * HID) + 2 * HID + col];
        const float hp  = hin[(long)m * HID + col];
        const float r   = sigmoidf_(ir + ar[j]);
        const float z   = sigmoidf_(iz + az[j]);
        const float n   = tanhf(inn + r * an[j]);
        const float hn  = (1.0f - z) * n + z * hp;
        hout[(long)m * HID + col] = hn;
        outt[(long)m * HID + col] = hn;
        if (hlast) hlast[(long)m * HID + col] = hn;
    }
}

// ---------------------------------------------------------------------------
// Row softmax, in place. One block per row (B*T rows of length S).
// ---------------------------------------------------------------------------
__global__ __launch_bounds__(256)
void softmax_rows(float* __restrict__ p, int S)
{
    __shared__ float red[256];
    float* row = p + (long)blockIdx.x * S;
    const int tid = threadIdx.x;

    float m = -INFINITY;
    for (int i = tid; i < S; i += 256) m = fmaxf(m, row[i]);
    red[tid] = m; __syncthreads();
    for (int s = 128; s > 0; s >>= 1) { if (tid < s) red[tid] = fmaxf(red[tid], red[tid + s]); __syncthreads(); }
    m = red[0]; __syncthreads();

    float sum = 0.0f;
    for (int i = tid; i < S; i += 256) { const float e = expf(row[i] - m); row[i] = e; sum += e; }
    red[tid] = sum; __syncthreads();
    for (int s = 128; s > 0; s >>= 1) { if (tid < s) red[tid] += red[tid + s]; __syncthreads(); }
    const float inv = 1.0f / red[0];

    for (int i = tid; i < S; i += 256) row[i] *= inv;
}

// ---------------------------------------------------------------------------
// LayerNorm over last dim (H=512). One block (256 thr, 2 elem/thr) per row.
// ---------------------------------------------------------------------------
__global__ __launch_bounds__(256)
void layernorm_rows(const float* __restrict__ x, const float* __restrict__ g,
                    const float* __restrict__ b, float* __restrict__ y)
{
    __shared__ float red[256];
    const long r = blockIdx.x;
    const float* row = x + r * HID;
    float* orow = y + r * HID;
    const int tid = threadIdx.x;

    const float a0 = row[tid], a1 = row[tid + 256];
    red[tid] = a0 + a1; __syncthreads();
    for (int s = 128; s > 0; s >>= 1) { if (tid < s) red[tid] += red[tid + s]; __syncthreads(); }
    const float mu = red[0] * (1.0f / HID);
    __syncthreads();

    const float d0 = a0 - mu, d1 = a1 - mu;
    red[tid] = d0 * d0 + d1 * d1; __syncthreads();
    for (int s = 128; s > 0; s >>= 1) { if (tid < s) red[tid] += red[tid + s]; __syncthreads(); }
    const float rstd = rsqrtf(red[0] * (1.0f / HID) + 1e-5f);

    orow[tid]       = d0 * rstd * g[tid]       + b[tid];
    orow[tid + 256] = d1 * rstd * g[tid + 256] + b[tid + 256];
}

// ---------------------------------------------------------------------------
// Small copies
// ---------------------------------------------------------------------------
__global__ void copy_f32(float* __restrict__ dst, const float* __restrict__ src, int n)
{
    const int i = blockIdx.x * blockDim.x + threadIdx.x;
    if (i < n) dst[i] = src[i];
}

// cat[(row)*1024 + 512 + col] = gout[row*512 + col]
__global__ void copy_out_to_cat(float* __restrict__ cat, const float* __restrict__ gout, int n)
{
    const int i = blockIdx.x * blockDim.x + threadIdx.x;
    if (i < n) {
        const int r = i >> 9;
        const int c = i & 511;
        cat[(long)r * (2 * HID) + HID + c] = gout[i];
    }
}

// ---------------------------------------------------------------------------
extern "C" void kernel_launch(void* const* d_in, const int* in_sizes, int n_in,
                              void* d_out, int out_size, void* d_ws, size_t ws_size,
                              hipStream_t stream)
{
    const float* input   = (const float*)d_in[0];   // [T,B,I]
    const float* hidden  = (const float*)d_in[1];   // [1,B,H]
    const float* enc     = (const float*)d_in[2];   // [S,B,2H]
    const float* lin_W   = (const float*)d_in[3];
    const float* lin_b   = (const float*)d_in[4];
    const float* w_ih    = (const float*)d_in[5];
    const float* w_hh    = (const float*)d_in[6];
    const float* b_ih    = (const float*)d_in[7];
    const float* b_hh    = (const float*)d_in[8];
    const float* attn_W  = (const float*)d_in[9];
    const float* attn_b  = (const float*)d_in[10];
    const float* attn2_W = (const float*)d_in[11];
    const float* attn2_b = (const float*)d_in[12];
    const float* comb_W  = (const float*)d_in[13];
    const float* comb_b  = (const float*)d_in[14];
    const float* ln_g    = (const float*)d_in[15];
    const float* ln_b    = (const float*)d_in[16];

    const int TB = TT * BBATCH;                 // 1024 flattened (t,b) rows

    // Workspace carve-up (floats)
    float* ws     = (float*)d_ws;
    float* x      = ws;                                   // [TB, H]   (later reused for Q)
    float* gi     = x      + (long)TB * HID;              // [TB, 3H]
    float* hping  = gi     + (long)TB * 3 * HID;          // [B, H]
    float* hpong  = hping  + (long)BBATCH * HID;          // [B, H]
    float* gout   = hpong  + (long)BBATCH * HID;          // [TB, H]  GRU outputs
    float* attenc = gout   + (long)TB * HID;              // [S*B, H]
    float* cat    = attenc + (long)SSRC * BBATCH * HID;   // [TB, 2H]
    float* comb   = cat    + (long)TB * 2 * HID;          // [TB, H]

    // Output carve-up: output | hidden_out | att_weight
    float* outp   = (float*)d_out;                        // [T,B,H]
    float* hidout = outp   + (long)TB * HID;              // [B,H]
    float* attw   = hidout + (long)BBATCH * HID;          // [B,T,S]

    const dim3 blk(128);   // 4 waves; block tile = 16 x 256

    // 1. x = input @ lin_W.T + lin_b           [TB,H] K=I
    gemm_wmma_f32<true><<<dim3(HID / 256, TB / 16, 1), blk, 0, stream>>>(
        input, IIN, 0, lin_W, IIN, 0, lin_b, x, HID, 0, TB, HID, IIN);

    // 2. gi = x @ w_ih.T + b_ih (all timesteps hoisted)   [TB,3H] K=H
    gemm_wmma_f32<true><<<dim3(3 * HID / 256, TB / 16, 1), blk, 0, stream>>>(
        x, HID, 0, w_ih, HID, 0, b_ih, gi, 3 * HID, 0, TB, 3 * HID, HID);

    // 3. att_enc = enc @ attn_W.T + attn_b     [S*B,H] K=2H  (dominant GEMM)
    gemm_wmma_f32<true><<<dim3(HID / 256, SSRC * BBATCH / 16, 1), blk, 0, stream>>>(
        enc, 2 * HID, 0, attn_W, 2 * HID, 0, attn_b, attenc, HID, 0,
        SSRC * BBATCH, HID, 2 * HID);

    // 4. h0 = hidden[0]
    copy_f32<<<(BBATCH * HID + 255) / 256, 256, 0, stream>>>(hping, hidden, BBATCH * HID);

    // 5. GRU recurrence: 32 sequential fused steps, ping-pong h
    float* hc = hping; float* hn = hpong;
    for (int t = 0; t < TT; ++t) {
        gru_step_wmma<<<dim3(HID / 16, BBATCH / 16), 32, 0, stream>>>(
            gi + (long)t * BBATCH * 3 * HID, hc, w_hh, b_hh, hn,
            gout + (long)t * BBATCH * HID,
            (t == TT - 1) ? hidout : nullptr);
        float* tmp = hc; hc = hn; hn = tmp;
    }

    // 6. Q = out @ attn2_W.T + attn2_b   (reuse x buffer)
    gemm_wmma_f32<true><<<dim3(HID / 256, TB / 16, 1), blk, 0, stream>>>(
        gout, HID, 0, attn2_W, HID, 0, attn2_b, x, HID, 0, TB, HID, HID);

    // 7. scores[b,t,s] = Q[t,b,:] . att_enc[s,b,:]  (batched strided NT GEMM)
    //    written straight into att_weight slot of d_out
    gemm_wmma_f32<true><<<dim3(SSRC / 256, TT / 16, BBATCH), blk, 0, stream>>>(
        x, (long)BBATCH * HID, HID,
        attenc, (long)BBATCH * HID, HID,
        nullptr,
        attw, SSRC, (long)TT * SSRC,
        TT, SSRC, HID);

    // 8. softmax over S, in place
    softmax_rows<<<BBATCH * TT, 256, 0, stream>>>(attw, SSRC);

    // 9. att_ap = att_w @ att_enc  (batched NN GEMM) -> cat[:, :H]
    gemm_wmma_f32<false><<<dim3(HID / 256, TT / 16, BBATCH), blk, 0, stream>>>(
        attw, SSRC, (long)TT * SSRC,
        attenc, (long)BBATCH * HID, HID,
        nullptr,
        cat, (long)BBATCH * 2 * HID, 2 * HID,
        TT, HID, SSRC);

    // 10. cat[:, H:] = out
    copy_out_to_cat<<<(TB * HID + 255) / 256, 256, 0, stream>>>(cat, gout, TB * HID);

    // 11. combined = cat @ comb_W.T + comb_b
    gemm_wmma_f32<true><<<dim3(HID / 256, TB / 16, 1), blk, 0, stream>>>(
        cat, 2 * HID, 0, comb_W, 2 * HID, 0, comb_b, comb, HID, 0, TB, HID, 2 * HID);

    // 12. output = layernorm(combined)
    layernorm_rows<<<TB, 256, 0, stream>>>(comb, ln_g, ln_b, outp);
}